// VPINN_46316927320560
// MI455X (gfx1250) — compile-verified
//
#include <hip/hip_runtime.h>
#include <math.h>

typedef __attribute__((ext_vector_type(2))) float v2f;
typedef __attribute__((ext_vector_type(8))) float v8f;

#define QPTS 4900
#define NEX 8
#define NEY 8
#define HID 64
#define TXN 10
#define TYN 10
#define NBP 16384
#define NPTS (NEX * NEY * QPTS)   /* 313600 interior points */
#define NTILES (NPTS / 16)        /* 19600 16-point tiles   */
#define NBTILES (NBP / 16)        /* 1024 boundary tiles    */

static __device__ inline v8f wmma_f32_16x16x4(v2f a, v2f b, v8f c) {
  // D = A(16x4,f32) * B(4x16,f32) + C(16x16,f32), full-wave WMMA
  return __builtin_amdgcn_wmma_f32_16x16x4_f32(
      /*neg_a=*/false, a, /*neg_b=*/false, b,
      /*c_mod=*/(short)0, c, /*reuse_a=*/false, /*reuse_b=*/false);
}

// ---------------------------------------------------------------------------
// Kernel 1: Laplacian of the MLP at all element-mapped quadrature points.
// 2 waves / block, each wave owns a 16-point tile and a private LDS act slice.
// Channels: 0=u, 1=u_x, 2=u_y, 3=u_xx, 4=u_yy.
// ---------------------------------------------------------------------------
__global__ __launch_bounds__(64) void vpinn_interior(
    const float* __restrict__ W1, const float* __restrict__ b1,
    const float* __restrict__ W2, const float* __restrict__ b2,
    const float* __restrict__ W3, const float* __restrict__ b3,
    const float* __restrict__ W4,
    const float* __restrict__ x_quad, const float* __restrict__ y_quad,
    const float* __restrict__ grid_x, const float* __restrict__ grid_y,
    float* __restrict__ lap_out)
{
  __shared__ float act[2][5][16 * HID];   // 40 KB
  __shared__ float xs[2][16], ys[2][16];

  const int wave = threadIdx.x >> 5;
  const int lane = threadIdx.x & 31;
  const int tile = blockIdx.x * 2 + wave;
  const int base = tile * 16;

  float* aV   = act[wave][0];
  float* aDX  = act[wave][1];
  float* aDY  = act[wave][2];
  float* aDXX = act[wave][3];
  float* aDYY = act[wave][4];

  // map 16 flat point ids -> physical (X,Y); flat index = (ex*NEY+ey)*Q + q
  if (lane < 16) {
    int g  = base + lane;
    int e  = g / QPTS;
    int q  = g - e * QPTS;
    int ex = e >> 3, ey = e & 7;
    float gx0 = grid_x[ex], gy0 = grid_y[ey];
    float jx = (grid_x[ex + 1] - gx0) * 0.5f;
    float jy = (grid_y[ey + 1] - gy0) * 0.5f;
    xs[wave][lane] = gx0 + jx * (x_quad[q] + 1.0f);
    ys[wave][lane] = gy0 + jy * (y_quad[q] + 1.0f);
  }
  // same-wave LDS, in-order: no barrier needed

  // ---- input layer: analytic derivatives in physical coords ----
  for (int i = lane; i < 16 * HID; i += 32) {
    int p = i >> 6, f = i & 63;
    float X = xs[wave][p], Y = ys[wave][p];
    float wx = W1[f], wy = W1[HID + f];
    float z = X * wx + Y * wy + b1[f];
    float h = tanhf(z);
    float hp  = 1.0f - h * h;
    float hpp = -2.0f * h * hp;
    aV[i]   = h;
    aDX[i]  = hp * wx;
    aDY[i]  = hp * wy;
    aDXX[i] = hpp * wx * wx;
    aDYY[i] = hpp * wy * wy;
  }

  const int m    = lane & 15;
  const int koff = (lane < 16) ? 0 : 2;
  const int phalf = (lane < 16) ? 0 : 8;

  // ---- hidden layers 2 and 3 via f32 WMMA ----
  for (int layer = 0; layer < 2; ++layer) {
    const float* __restrict__ W    = layer ? W3 : W2;
    const float* __restrict__ bias = layer ? b3 : b2;

    v8f acc[4][5];
#pragma unroll
    for (int nt = 0; nt < 4; ++nt) {
      float bb = bias[nt * 16 + m];
#pragma unroll
      for (int r = 0; r < 8; ++r) acc[nt][0][r] = bb;
#pragma unroll
      for (int ch = 1; ch < 5; ++ch)
#pragma unroll
        for (int r = 0; r < 8; ++r) acc[nt][ch][r] = 0.0f;
    }

    for (int k = 0; k < 16; ++k) {
      const int col = k * 4 + koff;
      v2f a[5];
#pragma unroll
      for (int ch = 0; ch < 5; ++ch)
        a[ch] = *reinterpret_cast<const v2f*>(&act[wave][ch][m * HID + col]);
#pragma unroll
      for (int nt = 0; nt < 4; ++nt) {
        const int n = nt * 16 + m;
        v2f bv;
        bv[0] = W[col * HID + n];
        bv[1] = W[(col + 1) * HID + n];
#pragma unroll
        for (int ch = 0; ch < 5; ++ch)
          acc[nt][ch] = wmma_f32_16x16x4(a[ch], bv, acc[nt][ch]);
      }
    }

    // tanh + derivative propagation, write back into the per-wave act slice
#pragma unroll
    for (int nt = 0; nt < 4; ++nt) {
#pragma unroll
      for (int r = 0; r < 8; ++r) {
        int pt  = r + phalf;
        int idx = pt * HID + nt * 16 + m;
        float h   = tanhf(acc[nt][0][r]);
        float hp  = 1.0f - h * h;
        float hpp = -2.0f * h * hp;
        float gx = acc[nt][1][r], gy = acc[nt][2][r];
        aV[idx]   = h;
        aDX[idx]  = hp * gx;
        aDY[idx]  = hp * gy;
        aDXX[idx] = hpp * gx * gx + hp * acc[nt][3][r];
        aDYY[idx] = hpp * gy * gy + hp * acc[nt][4][r];
      }
    }
  }

  // ---- linear output layer: lap = (h_xx + h_yy) . W4 (b4 drops out) ----
  if (lane < 16) {
    float s = 0.0f;
#pragma unroll 8
    for (int f = 0; f < HID; ++f)
      s += (aDXX[lane * HID + f] + aDYY[lane * HID + f]) * W4[f];
    lap_out[base + lane] = s;
  }
}

// ---------------------------------------------------------------------------
// Kernel 2: boundary MLP forward (value channel only) -> per-tile SSE partials
// 4 waves / block, 16 points / wave.
// ---------------------------------------------------------------------------
__global__ __launch_bounds__(128) void vpinn_boundary(
    const float* __restrict__ W1, const float* __restrict__ b1,
    const float* __restrict__ W2, const float* __restrict__ b2,
    const float* __restrict__ W3, const float* __restrict__ b3,
    const float* __restrict__ W4, const float* __restrict__ b4,
    const float* __restrict__ boundary_xy, const float* __restrict__ u_bound,
    float* __restrict__ bpart)
{
  __shared__ float act[4][16 * HID];   // 16 KB
  __shared__ float sq[4][16];

  const int wave = threadIdx.x >> 5;
  const int lane = threadIdx.x & 31;
  const int tile = blockIdx.x * 4 + wave;
  const int base = tile * 16;
  float* A = act[wave];

  for (int i = lane; i < 16 * HID; i += 32) {
    int p = i >> 6, f = i & 63;
    float X = boundary_xy[(base + p) * 2 + 0];
    float Y = boundary_xy[(base + p) * 2 + 1];
    A[i] = tanhf(X * W1[f] + Y * W1[HID + f] + b1[f]);
  }

  const int m    = lane & 15;
  const int koff = (lane < 16) ? 0 : 2;
  const int phalf = (lane < 16) ? 0 : 8;

  for (int layer = 0; layer < 2; ++layer) {
    const float* __restrict__ W    = layer ? W3 : W2;
    const float* __restrict__ bias = layer ? b3 : b2;
    v8f acc[4];
#pragma unroll
    for (int nt = 0; nt < 4; ++nt) {
      float bb = bias[nt * 16 + m];
#pragma unroll
      for (int r = 0; r < 8; ++r) acc[nt][r] = bb;
    }
    for (int k = 0; k < 16; ++k) {
      const int col = k * 4 + koff;
      v2f a = *reinterpret_cast<const v2f*>(&A[m * HID + col]);
#pragma unroll
      for (int nt = 0; nt < 4; ++nt) {
        const int n = nt * 16 + m;
        v2f bv;
        bv[0] = W[col * HID + n];
        bv[1] = W[(col + 1) * HID + n];
        acc[nt] = wmma_f32_16x16x4(a, bv, acc[nt]);
      }
    }
#pragma unroll
    for (int nt = 0; nt < 4; ++nt)
#pragma unroll
      for (int r = 0; r < 8; ++r)
        A[(r + phalf) * HID + nt * 16 + m] = tanhf(acc[nt][r]);
  }

  if (lane < 16) {
    float u = b4[0];
#pragma unroll 8
    for (int f = 0; f < HID; ++f) u += A[lane * HID + f] * W4[f];
    float d = u - u_bound[base + lane];
    sq[wave][lane] = d * d;
  }
  if (lane == 0) {
    float s = 0.0f;
#pragma unroll
    for (int j = 0; j < 16; ++j) s += sq[wave][j];
    bpart[tile] = s;   // deterministic partial (no float atomics)
  }
}

// ---------------------------------------------------------------------------
// Kernel 3: per-element quadrature einsum + residual -> mean-square per element
// ---------------------------------------------------------------------------
__global__ __launch_bounds__(128) void vpinn_element(
    const float* __restrict__ lap, const float* __restrict__ w_quad,
    const float* __restrict__ v_x, const float* __restrict__ v_y,
    const float* __restrict__ F_ext,
    const float* __restrict__ grid_x, const float* __restrict__ grid_y,
    float* __restrict__ eloss)
{
  __shared__ float red[128];
  const int el = blockIdx.x;
  const int ex = el >> 3, ey = el & 7;
  const int t = threadIdx.x;

  float v = 0.0f;
  if (t < TYN * TXN) {
    int kk = t / TXN, r = t - kk * TXN;
    const float* lp  = lap + el * QPTS;
    const float* vxr = v_x + r * QPTS;
    const float* vyk = v_y + kk * QPTS;
    float s = 0.0f;
    for (int q = 0; q < QPTS; ++q)
      s += w_quad[2 * q] * w_quad[2 * q + 1] * vxr[q] * vyk[q] * lp[q];
    float jx = (grid_x[ex + 1] - grid_x[ex]) * 0.5f;
    float jy = (grid_y[ey + 1] - grid_y[ey]) * 0.5f;
    float res = jx * jy * s - F_ext[(el * TYN + kk) * TXN + r];
    v = res * res;
  }
  red[t] = v;
  __syncthreads();
  for (int off = 64; off > 0; off >>= 1) {
    if (t < off) red[t] += red[t + off];
    __syncthreads();
  }
  if (t == 0) eloss[el] = red[0] / (float)(TYN * TXN);
}

// ---------------------------------------------------------------------------
// Kernel 4: final deterministic reduction -> scalar loss
// ---------------------------------------------------------------------------
__global__ __launch_bounds__(256) void vpinn_final(
    const float* __restrict__ bpart, const float* __restrict__ eloss,
    float* __restrict__ out)
{
  __shared__ float rb[256], rv[256];
  const int t = threadIdx.x;
  float sb = 0.0f, sv = 0.0f;
  for (int i = t; i < NBTILES; i += 256) sb += bpart[i];
  for (int i = t; i < NEX * NEY; i += 256) sv += eloss[i];
  rb[t] = sb; rv[t] = sv;
  __syncthreads();
  for (int off = 128; off > 0; off >>= 1) {
    if (t < off) { rb[t] += rb[t + off]; rv[t] += rv[t + off]; }
    __syncthreads();
  }
  if (t == 0) out[0] = 10.0f * (rb[0] / (float)NBP) + rv[0];
}

extern "C" void kernel_launch(void* const* d_in, const int* in_sizes, int n_in,
                              void* d_out, int out_size, void* d_ws, size_t ws_size,
                              hipStream_t stream) {
  const float* W1 = (const float*)d_in[0];
  const float* b1 = (const float*)d_in[1];
  const float* W2 = (const float*)d_in[2];
  const float* b2 = (const float*)d_in[3];
  const float* W3 = (const float*)d_in[4];
  const float* b3 = (const float*)d_in[5];
  const float* W4 = (const float*)d_in[6];
  const float* b4 = (const float*)d_in[7];
  const float* x_quad  = (const float*)d_in[8];
  const float* y_quad  = (const float*)d_in[9];
  const float* w_quad  = (const float*)d_in[10];
  const float* v_x     = (const float*)d_in[11];
  const float* v_y     = (const float*)d_in[12];
  const float* F_ext   = (const float*)d_in[13];
  const float* bxy     = (const float*)d_in[14];
  const float* ub      = (const float*)d_in[15];
  const float* grid_x  = (const float*)d_in[16];
  const float* grid_y  = (const float*)d_in[17];

  float* ws    = (float*)d_ws;
  float* lap   = ws;                 // NPTS floats
  float* bpart = ws + NPTS;          // NBTILES floats
  float* eloss = bpart + NBTILES;    // 64 floats

  vpinn_interior<<<NTILES / 2, 64, 0, stream>>>(
      W1, b1, W2, b2, W3, b3, W4, x_quad, y_quad, grid_x, grid_y, lap);
  vpinn_boundary<<<NBTILES / 4, 128, 0, stream>>>(
      W1, b1, W2, b2, W3, b3, W4, b4, bxy, ub, bpart);
  vpinn_element<<<NEX * NEY, 128, 0, stream>>>(
      lap, w_quad, v_x, v_y, F_ext, grid_x, grid_y, eloss);
  vpinn_final<<<1, 256, 0, stream>>>(bpart, eloss, (float*)d_out);
}